// NeuronClass_1219770712823
// MI455X (gfx1250) — compile-verified
//
#include <hip/hip_runtime.h>
#include <stdint.h>

typedef __attribute__((ext_vector_type(2))) float v2f;
typedef __attribute__((ext_vector_type(8))) float v8f;

#define ALPHA 0.995f
#define VTH   2.0f
#define NN    512
#define NP    516        // padded LDS row stride in floats (bank-conflict-free, 16B aligned)
#define KC    16         // timesteps per chunk = one WMMA t-tile
#define NCH   125        // 2000 / 16, uniform chunks

#if defined(__gfx1250__)
#define ASYNC_WAIT(n) asm volatile("s_wait_asynccnt " #n ::: "memory")
#else
#define ASYNC_WAIT(n)
#endif

// Stage one KC x NN chunk of x into LDS. On gfx1250: async DMA to LDS
// (GLOBAL_LOAD_ASYNC_TO_LDS_B128, GVS mode, tracked by ASYNCcnt).
__device__ __forceinline__ void issue_chunk(const float* __restrict__ src,
                                            float* __restrict__ dst, int tid) {
#if defined(__gfx1250__)
    const unsigned long long base = (unsigned long long)(uintptr_t)src;
    #pragma unroll
    for (int j = 0; j < 4; ++j) {
        const int i4 = tid + j * 512;          // float4 index within chunk
        const int tt = i4 >> 7;                // / (NN/4)
        const int nn = (i4 & 127) << 2;
        const unsigned glOff = (unsigned)(i4 << 4);                      // bytes
        const unsigned ldsA  = (unsigned)(uintptr_t)(dst + tt * NP + nn); // LDS byte addr
        asm volatile("global_load_async_to_lds_b128 %0, %1, %2"
                     :: "v"(ldsA), "v"(glOff), "s"(base) : "memory");
    }
#else
    for (int j = 0; j < 4; ++j) {
        const int i4 = tid + j * 512;
        const int tt = i4 >> 7;
        const int nn = (i4 & 127) << 2;
        *(float4*)(dst + tt * NP + nn) = *(const float4*)(src + (size_t)i4 * 4);
    }
#endif
}

// LDS layout (floats): xl0[KC*NP] | xl1[KC*NP] | wl[512] | sp[16*16] | sl/vl/cl/zl[KC each]
extern "C" __global__ __launch_bounds__(512)
void lif_eprop_kernel(const float* __restrict__ x, const float* __restrict__ w,
                      float* __restrict__ vout, float* __restrict__ zout,
                      float* __restrict__ gout, int T)
{
    extern __shared__ float smem[];
    float* xl0 = smem;
    float* xl1 = xl0 + KC * NP;
    float* wl  = xl1 + KC * NP;        // 512
    float* sp  = wl + NN;              // 256 (16 waves x 16 t)
    float* sl  = sp + 256;             // KC
    float* vl  = sl + KC;              // KC
    float* cl  = vl + KC;              // KC
    float* zl  = cl + KC;              // KC

    const int tid  = threadIdx.x;
    const int b    = blockIdx.x;
    const int lane = tid & 31;
    const int wv   = tid >> 5;

    wl[tid] = w[tid];
    __syncthreads();

    // wsq = ||w||^2 (redundant, LDS-broadcast reads, once per block)
    float wsq = 0.f;
    for (int i = 0; i < NN; ++i) { float wi = wl[i]; wsq = fmaf(wi, wi, wsq); }

    const float wcol = wl[tid];
    float p = 0.f, g = 0.f, vcar = 0.f;

    const float* xb = x + (size_t)b * (size_t)T * NN;

    // prologue: chunk 0 in flight
    issue_chunk(xb, xl0, tid);

    for (int c = 0; c < NCH; ++c) {
        float* xcur = (c & 1) ? xl1 : xl0;
        float* xnxt = (c & 1) ? xl0 : xl1;

        if (c + 1 < NCH) {
            issue_chunk(xb + (size_t)(c + 1) * KC * NN, xnxt, tid);
            ASYNC_WAIT(0x4);     // next chunk's 4 may stay in flight; current chunk done
        } else {
            ASYNC_WAIT(0x0);
        }
        __syncthreads();         // all waves' async data visible

        // ---- WMMA partial dots: s[t] = x_t . w  (V_WMMA_F32_16X16X4_F32) ----
        // A[m,k] = w[k0+k] broadcast over rows; B[k,j] = x[j][k0+k]
        // => column j of D = partial dot for local timestep j; lane j reads acc[0].
        {
            const int n0  = wv << 5;                 // 32-wide N slice per wave
            const int kh  = (lane < 16) ? 0 : 2;     // K offset for this VGPR half
            const int trw = lane & 15;               // timestep row for B
            v8f acc = {};
            #pragma unroll
            for (int k = n0; k < n0 + 32; k += 4) {
                v2f a, bm;
                a.x  = wl[k + kh];
                a.y  = wl[k + kh + 1];
                bm.x = xcur[trw * NP + k + kh];
                bm.y = xcur[trw * NP + k + kh + 1];
                acc = __builtin_amdgcn_wmma_f32_16x16x4_f32(
                          false, a, false, bm, (short)0, acc, false, false);
            }
            if (lane < 16) sp[wv * 16 + lane] = acc[0];
        }
        __syncthreads();

        // ---- reduce 16 N-slice partials per timestep ----
        if (tid < KC) {
            float s = 0.f;
            #pragma unroll
            for (int q = 0; q < 16; ++q) s += sp[q * 16 + tid];
            sl[tid] = s;
        }
        __syncthreads();

        // ---- serial leak scan (thread 0), broadcast via LDS ----
        if (tid == 0) {
            float v = vcar;
            #pragma unroll
            for (int t = 0; t < KC; ++t) {
                float s = sl[t];
                v = fmaf(ALPHA, v, s);
                vl[t] = v;
                zl[t] = (v - VTH > 0.f) ? 1.f : 0.f;
                cl[t] = fmaf(-v, wsq, s);        // eps @ w = s - v*||w||^2
            }
            vcar = v;
        }
        __syncthreads();

        // ---- emit v_seq, z_seq ----
        if (tid < KC) {
            vout[(size_t)b * T + c * KC + tid] = vl[tid];
            zout[(size_t)b * T + c * KC + tid] = zl[tid];
        }

        // ---- elementwise eligibility update, one column per thread ----
        #pragma unroll
        for (int t = 0; t < KC; ++t) {
            float v  = vl[t];
            float cc = cl[t];
            float xt = xcur[t * NP + tid];
            float eps = fmaf(-v, wcol, xt);      // x_t - v*w
            g = fmaf(v, eps, g);
            g = fmaf(cc, p, g);
            p = fmaf(ALPHA, p, xt);
        }
        __syncthreads();   // buffer xcur is rewritten by the next iteration's prefetch
    }

    gout[(size_t)b * NN + tid] = g;
}

extern "C" void kernel_launch(void* const* d_in, const int* in_sizes, int n_in,
                              void* d_out, int out_size, void* d_ws, size_t ws_size,
                              hipStream_t stream) {
    const float* x = (const float*)d_in[0];
    const float* w = (const float*)d_in[1];
    const int B = 128, T = 2000;

    float* vout = (float*)d_out;
    float* zout = vout + (size_t)B * T;
    float* gout = zout + (size_t)B * T;

    const size_t shmem = (size_t)(2 * KC * NP + NN + 256 + 4 * KC) * sizeof(float);
    hipFuncSetAttribute((const void*)lif_eprop_kernel,
                        hipFuncAttributeMaxDynamicSharedMemorySize, (int)shmem);
    lif_eprop_kernel<<<dim3(B), dim3(512), shmem, stream>>>(x, w, vout, zout, gout, T);
}